// NonlocalInteraction_65377992180231
// MI455X (gfx1250) — compile-verified
//
#include <hip/hip_runtime.h>
#include <hip/hip_bf16.h>

// ---------------------------------------------------------------------------
// NonlocalInteraction (SpookyNet-style) for MI455X / gfx1250, wave32 + WMMA.
//   N = 8192 rows, D = 512 features, NB = 32 molecules (block-diag attention).
// Full-fp32 matrix math via V_WMMA_F32_16X16X4_F32 (D = A(16x4) x B(4x16) + C).
// d_ws requirement: 6 * N * D * 4 bytes = 96 MB.
// ---------------------------------------------------------------------------

typedef __attribute__((ext_vector_type(2))) float v2f;   // A/B fragment (16x4 / 4x16 f32)
typedef __attribute__((ext_vector_type(8))) float v8f;   // C/D fragment (16x16 f32)

#define DD   512
#define NROW 8192

__device__ __forceinline__ float swishf(float x) {
    // alpha = 1.0, beta = 1.702  (frozen learnable swish)
    return x * (1.0f / (1.0f + __expf(-1.702f * x)));
}

__global__ __launch_bounds__(256) void swish_kernel(const float* __restrict__ in,
                                                    float* __restrict__ outp, int n) {
    int i = blockIdx.x * 256 + threadIdx.x;
    if (i < n) outp[i] = swishf(in[i]);
}

// ---------------------------------------------------------------------------
// C[M,512] = epilogue( A[M,512] @ W^T + bias )
//   MODE 0: out = swish(acc + bias)                      (layer 1)
//   MODE 1: out = swish(R + acc + bias)   (residual h)   (layer 2)
//   MODE 2: out = acc + bias                             (output linear)
// Block tile 128(M) x 64(N), BK=16, 256 threads = 8 waves, wave tile 32x32.
// ---------------------------------------------------------------------------
template<int MODE>
__global__ __launch_bounds__(256) void mlp_gemm(const float* __restrict__ A,
                                                const float* __restrict__ W,
                                                const float* __restrict__ bias,
                                                const float* __restrict__ R,
                                                float* __restrict__ C) {
    constexpr int BM = 128, BN = 64, BK = 16;
    __shared__ float As[BK][BM + 8];   // +8 pad: half0/half1 fragment reads hit disjoint banks
    __shared__ float Bs[BK][BN + 8];

    const int m0   = blockIdx.x * BM;
    const int n0   = blockIdx.y * BN;
    const int t    = threadIdx.x;
    const int wave = t >> 5;
    const int lane = t & 31;
    const int half = lane >> 4;        // 0: K0/K1 rows, 1: K2/K3 rows
    const int l    = lane & 15;
    const int wm   = (wave & 3) * 32;  // wave tile origin inside block tile
    const int wn   = (wave >> 2) * 32;

    const int lrow = t >> 2;           // 0..63 (loader row)
    const int lc4  = (t & 3) * 4;      // 0,4,8,12 (loader col group)

    v8f acc[2][2] = {};

    for (int k0 = 0; k0 < DD; k0 += BK) {
        // stage A (128x16, transposed to k-major) and W (64 rows n, 16 cols k)
        float4 fa0 = *(const float4*)(A + (size_t)(m0 + lrow) * DD + k0 + lc4);
        float4 fa1 = *(const float4*)(A + (size_t)(m0 + lrow + 64) * DD + k0 + lc4);
        float4 fb  = *(const float4*)(W + (size_t)(n0 + lrow) * DD + k0 + lc4);
        As[lc4 + 0][lrow] = fa0.x; As[lc4 + 1][lrow] = fa0.y;
        As[lc4 + 2][lrow] = fa0.z; As[lc4 + 3][lrow] = fa0.w;
        As[lc4 + 0][lrow + 64] = fa1.x; As[lc4 + 1][lrow + 64] = fa1.y;
        As[lc4 + 2][lrow + 64] = fa1.z; As[lc4 + 3][lrow + 64] = fa1.w;
        Bs[lc4 + 0][lrow] = fb.x; Bs[lc4 + 1][lrow] = fb.y;
        Bs[lc4 + 2][lrow] = fb.z; Bs[lc4 + 3][lrow] = fb.w;
        __syncthreads();

        #pragma unroll
        for (int kk = 0; kk < BK; kk += 4) {
            v2f a[2], b[2];
            #pragma unroll
            for (int i = 0; i < 2; ++i) {   // A frag: lane = M, half selects K pair
                a[i].x = As[kk + half * 2 + 0][wm + i * 16 + l];
                a[i].y = As[kk + half * 2 + 1][wm + i * 16 + l];
            }
            #pragma unroll
            for (int j = 0; j < 2; ++j) {   // B frag: lane = N, half selects K pair
                b[j].x = Bs[kk + half * 2 + 0][wn + j * 16 + l];
                b[j].y = Bs[kk + half * 2 + 1][wn + j * 16 + l];
            }
            #pragma unroll
            for (int i = 0; i < 2; ++i)
                #pragma unroll
                for (int j = 0; j < 2; ++j)
                    acc[i][j] = __builtin_amdgcn_wmma_f32_16x16x4_f32(
                        false, a[i], false, b[j], (short)0, acc[i][j], false, false);
        }
        __syncthreads();
    }

    // epilogue: C/D layout -> vgpr v holds row v + 8*half, lane l holds col
    #pragma unroll
    for (int i = 0; i < 2; ++i) {
        #pragma unroll
        for (int j = 0; j < 2; ++j) {
            const int col = n0 + wn + j * 16 + l;
            const float bv = bias[col];
            #pragma unroll
            for (int vv = 0; vv < 8; ++vv) {
                const int row = m0 + wm + i * 16 + vv + half * 8;
                float val = acc[i][j][vv] + bv;
                if (MODE == 1) val += R[(size_t)row * DD + col];
                if (MODE != 2) val = swishf(val);
                C[(size_t)row * DD + col] = val;
            }
        }
    }
}

// ---------------------------------------------------------------------------
// Flash attention over the block-diagonal mask. One workgroup = 16 query rows.
// batch_seg is sorted, so the allowed keys for a query tile are a contiguous
// range found by binary search. 8 waves each own a 64-wide D slice; partial
// S (over the slice) is reduced in LDS, masked online-softmax'd, then P@V is
// accumulated per-slice with WMMA.
// ---------------------------------------------------------------------------
__global__ __launch_bounds__(256) void attn_kernel(const float* __restrict__ q,
                                                   const float* __restrict__ k,
                                                   const float* __restrict__ v,
                                                   const int* __restrict__ seg,
                                                   float* __restrict__ outp) {
    constexpr int KT = 64;               // keys per tile
    __shared__ float Sbuf[8][16][68];    // per-wave partial S (padded)
    __shared__ float Ps[16][68];         // reduced S -> P (padded)
    __shared__ float mrow[16], lrowS[16], arow[16];
    __shared__ int   srange[2];
    __shared__ int   segq[16];

    const int q0   = blockIdx.x * 16;
    const int t    = threadIdx.x;
    const int wave = t >> 5;
    const int lane = t & 31;
    const int half = lane >> 4;
    const int l    = lane & 15;
    const int d0   = wave * 64;          // this wave's D slice

    if (t < 16) { mrow[t] = -1e30f; lrowS[t] = 0.0f; segq[t] = seg[q0 + t]; }
    if (t == 0) {
        const int slo = seg[q0], shi = seg[q0 + 15];
        int lo = 0, hi = NROW;           // lower_bound(seg, slo)
        while (lo < hi) { int m = (lo + hi) >> 1; if (seg[m] < slo) lo = m + 1; else hi = m; }
        srange[0] = lo;
        lo = 0; hi = NROW;               // upper_bound(seg, shi)
        while (lo < hi) { int m = (lo + hi) >> 1; if (seg[m] <= shi) lo = m + 1; else hi = m; }
        srange[1] = lo;
    }
    __syncthreads();
    const int klo = srange[0] & ~(KT - 1);
    const int khi = srange[1];

    // q fragments for this wave's slice, kept in registers (16 K-steps of 4)
    v2f qf[16];
    #pragma unroll
    for (int kk = 0; kk < 16; ++kk) {
        const float* qp = q + (size_t)(q0 + l) * DD + d0 + kk * 4 + half * 2;
        qf[kk].x = qp[0]; qf[kk].y = qp[1];
    }

    v8f oacc[4] = {};                       // O tile: 16 x 64 (4 column sub-tiles)
    const float rscale = 0.044194173824159216f;  // 1/sqrt(512)

    for (int kt = klo; kt < khi; kt += KT) {
        // ---- partial S = q_slice @ k_slice^T  (16 x 64 keys) ----
        v8f sacc[4] = {};
        #pragma unroll
        for (int kk = 0; kk < 16; ++kk) {
            #pragma unroll
            for (int j = 0; j < 4; ++j) {
                int key = kt + j * 16 + l;
                if (key >= NROW) key = NROW - 1;   // value irrelevant: masked below
                const float* kp = k + (size_t)key * DD + d0 + kk * 4 + half * 2;
                v2f b; b.x = kp[0]; b.y = kp[1];
                sacc[j] = __builtin_amdgcn_wmma_f32_16x16x4_f32(
                    false, qf[kk], false, b, (short)0, sacc[j], false, false);
            }
        }
        #pragma unroll
        for (int j = 0; j < 4; ++j)
            #pragma unroll
            for (int vv = 0; vv < 8; ++vv)
                Sbuf[wave][vv + half * 8][j * 16 + l] = sacc[j][vv];
        __syncthreads();

        // ---- reduce slices, scale, mask ----
        for (int e = t; e < 16 * 64; e += 256) {
            const int r = e >> 6, c = e & 63;
            float s = 0.f;
            #pragma unroll
            for (int w = 0; w < 8; ++w) s += Sbuf[w][r][c];
            const int key = kt + c;
            const bool ok = (key < NROW) && (seg[key] == segq[r]);
            Ps[r][c] = ok ? s * rscale : -1e30f;
        }
        __syncthreads();

        // ---- online softmax stats (16 rows, one thread each; tiny) ----
        if (t < 16) {
            const float mo = mrow[t];
            float tm = mo;
            for (int c = 0; c < 64; ++c) tm = fmaxf(tm, Ps[t][c]);
            const float al = __expf(mo - tm);   // ==0 when first real tile arrives
            float sum = 0.f;
            for (int c = 0; c < 64; ++c) {
                const float p = __expf(Ps[t][c] - tm);
                Ps[t][c] = p;
                sum += p;
            }
            mrow[t] = tm;
            lrowS[t] = lrowS[t] * al + sum;
            arow[t] = al;
        }
        __syncthreads();

        // ---- rescale running O, then O += P @ V_slice ----
        #pragma unroll
        for (int j = 0; j < 4; ++j)
            #pragma unroll
            for (int vv = 0; vv < 8; ++vv)
                oacc[j][vv] *= arow[vv + half * 8];

        #pragma unroll
        for (int kk = 0; kk < 16; ++kk) {
            v2f a;                                  // A frag of P: lane = query row
            a.x = Ps[l][kk * 4 + half * 2 + 0];
            a.y = Ps[l][kk * 4 + half * 2 + 1];
            int key0 = kt + kk * 4 + half * 2;
            int key1 = key0 + 1;
            if (key0 >= NROW) key0 = NROW - 1;      // masked keys have P == 0
            if (key1 >= NROW) key1 = NROW - 1;
            #pragma unroll
            for (int j = 0; j < 4; ++j) {
                v2f b;                              // B frag of V: lane = out column
                b.x = v[(size_t)key0 * DD + d0 + j * 16 + l];
                b.y = v[(size_t)key1 * DD + d0 + j * 16 + l];
                oacc[j] = __builtin_amdgcn_wmma_f32_16x16x4_f32(
                    false, a, false, b, (short)0, oacc[j], false, false);
            }
        }
        __syncthreads();   // protect Ps/Sbuf for next key tile
    }

    // ---- normalize and store ----
    #pragma unroll
    for (int j = 0; j < 4; ++j) {
        #pragma unroll
        for (int vv = 0; vv < 8; ++vv) {
            const int r = vv + half * 8;
            const float inv = 1.0f / lrowS[r];
            outp[(size_t)(q0 + r) * DD + d0 + j * 16 + l] = oacc[j][vv] * inv;
        }
    }
}

// ---------------------------------------------------------------------------
extern "C" void kernel_launch(void* const* d_in, const int* in_sizes, int n_in,
                              void* d_out, int out_size, void* d_ws, size_t ws_size,
                              hipStream_t stream) {
    (void)in_sizes; (void)n_in; (void)out_size; (void)ws_size;

    // setup_inputs order: [q_W1,q_b1,q_W2,q_b2,q_Wo,q_bo, k_*..., v_*..., x, batch_seg, num_batch]
    const float* x   = (const float*)d_in[18];
    const int*   seg = (const int*)d_in[19];

    float* ws = (float*)d_ws;
    const size_t ND = (size_t)NROW * DD;
    float* sx = ws;            // swish(x), shared by all branches
    float* t1 = ws + 1 * ND;   // swish(layer1)
    float* t2 = ws + 2 * ND;   // swish(x + layer2)
    float* qkv[3] = { ws + 3 * ND, ws + 4 * ND, ws + 5 * ND };

    swish_kernel<<<dim3((unsigned)((ND + 255) / 256)), dim3(256), 0, stream>>>(x, sx, (int)ND);

    const dim3 gg(NROW / 128, DD / 64), bb(256);
    for (int br = 0; br < 3; ++br) {
        const float* W1 = (const float*)d_in[br * 6 + 0];
        const float* b1 = (const float*)d_in[br * 6 + 1];
        const float* W2 = (const float*)d_in[br * 6 + 2];
        const float* b2 = (const float*)d_in[br * 6 + 3];
        const float* Wo = (const float*)d_in[br * 6 + 4];
        const float* bo = (const float*)d_in[br * 6 + 5];
        mlp_gemm<0><<<gg, bb, 0, stream>>>(sx, W1, b1, nullptr, t1);
        mlp_gemm<1><<<gg, bb, 0, stream>>>(t1, W2, b2, x, t2);
        mlp_gemm<2><<<gg, bb, 0, stream>>>(t2, Wo, bo, nullptr, qkv[br]);
    }

    attn_kernel<<<dim3(NROW / 16), dim3(256), 0, stream>>>(qkv[0], qkv[1], qkv[2], seg,
                                                           (float*)d_out);
}